// hv_mamba_54546084659536
// MI455X (gfx1250) — compile-verified
//
#include <hip/hip_runtime.h>

// Mamba 2D selective scan for MI455X (gfx1250, wave32).
// One workgroup per sequence; proj & delta GEMMs via V_WMMA_F32_16X16X4_F32;
// recurrent scan with per-thread register state; everything fused in LDS.

typedef float v2f __attribute__((ext_vector_type(2)));
typedef float v8f __attribute__((ext_vector_type(8)));

#define C_DIM   128   // channels d
#define L_DIM   128   // sequence length
#define N_STATE 16
#define RT      4     // DT_RANK
#define P_DIM   36    // RT + 2*N_STATE
#define U_STR   132   // LDS row stride for u tile (132%64==4 -> conflict-free frag reads)
#define P_STR   48    // padded proj width (3 x 16 col tiles)
#define HW      (128 * 128)

__launch_bounds__(256, 1)
__global__ void mamba_seq_kernel(const float* __restrict__ x,
                                 const float* __restrict__ A_log,   // [C][N]
                                 const float* __restrict__ Dvec,    // [C]
                                 const float* __restrict__ xpw,     // [36][C]
                                 const float* __restrict__ dtw,     // [C][4]
                                 const float* __restrict__ dtb,     // [C]
                                 float* __restrict__ out,
                                 int inner_mul,   // W for h-branch, 1 for v-branch
                                 int l_stride,    // 1 for h-branch, W for v-branch
                                 int accumulate)  // 0: overwrite, 1: +=
{
    __shared__ float u_s[L_DIM * U_STR];       // u tile, later reused to hold y
    __shared__ float delta_s[L_DIM * C_DIM];   // softplus output
    __shared__ float proj_s[L_DIM * P_STR];    // [dt_raw(0:4) | Bc(4:20) | Cc(20:36)]

    const int tid   = threadIdx.x;
    const int lane  = tid & 31;
    const int wv    = tid >> 5;          // 8 waves
    const int lrow  = lane & 15;         // M/N index inside a 16-wide tile
    const int khalf = lane >> 4;         // K half-select for 16x16x4 fragments

    const int s     = blockIdx.x;        // sequence id (B*128 of them)
    const int b     = s >> 7;
    const int inner = s & 127;
    const int base  = b * (C_DIM * HW) + inner * inner_mul;

    // ---- Phase 1: stage u[L][C] into LDS (lane-varying l => coalesced for h-branch,
    //      L2-resident strided for v-branch) ----
    for (int idx = tid; idx < L_DIM * C_DIM; idx += 256) {
        int c = idx >> 7, l = idx & 127;
        u_s[l * U_STR + c] = x[base + c * HW + l * l_stride];
    }
    __syncthreads();

    // ---- Phase 2: proj[l][p] = sum_c u[l][c] * xpw[p][c]  (fp32 WMMA, K=4 steps) ----
    for (int i = 0; i < 3; ++i) {
        int job = wv + 8 * i;            // 24 (mt,nt) tile jobs over 8 waves
        int mt = job / 3, nt = job % 3;
        int p  = nt * 16 + lrow;
        int pc = (p < P_DIM) ? p : 0;
        v8f acc = {0.f, 0.f, 0.f, 0.f, 0.f, 0.f, 0.f, 0.f};
        for (int kk = 0; kk < 32; ++kk) {
            int cb = kk * 4 + 2 * khalf;
            v2f a;
            a.x = u_s[(mt * 16 + lrow) * U_STR + cb];
            a.y = u_s[(mt * 16 + lrow) * U_STR + cb + 1];
            v2f bf = *(const v2f*)(xpw + pc * C_DIM + cb);
            if (p >= P_DIM) { bf.x = 0.f; bf.y = 0.f; }
            acc = __builtin_amdgcn_wmma_f32_16x16x4_f32(false, a, false, bf,
                                                        (short)0, acc, false, false);
        }
        #pragma unroll
        for (int j = 0; j < 8; ++j) {
            int row = mt * 16 + j + 8 * khalf;
            int col = nt * 16 + lrow;
            proj_s[row * P_STR + col] = acc[j];
        }
    }
    __syncthreads();

    // ---- Phase 3: delta[l][d] = softplus(sum_r dt_raw[l][r]*dtw[d][r] + dtb[d]),
    //      rank-4 contraction == exactly one 16x16x4 WMMA per tile ----
    {
        int nt = wv;                      // wave -> one 16-wide d tile, all 8 l tiles
        int d  = nt * 16 + lrow;
        int rb = 2 * khalf;
        v2f bf = *(const v2f*)(dtw + d * 4 + rb);   // constant across mt
        float bias = dtb[d];
        for (int mt = 0; mt < 8; ++mt) {
            v2f a;
            a.x = proj_s[(mt * 16 + lrow) * P_STR + rb];
            a.y = proj_s[(mt * 16 + lrow) * P_STR + rb + 1];
            v8f acc = {0.f, 0.f, 0.f, 0.f, 0.f, 0.f, 0.f, 0.f};
            acc = __builtin_amdgcn_wmma_f32_16x16x4_f32(false, a, false, bf,
                                                        (short)0, acc, false, false);
            #pragma unroll
            for (int j = 0; j < 8; ++j) {
                int l = mt * 16 + j + 8 * khalf;
                float v = acc[j] + bias;
                float sp = (v > 20.f) ? v : __logf(1.f + __expf(v));
                delta_s[l * C_DIM + d] = sp;
            }
        }
    }
    __syncthreads();

    // ---- Phase 4: recurrent scan. Thread pair (2t,2t+1) owns channel c = tid>>1,
    //      each half keeps 8 of the 16 states in registers. ----
    {
        int c  = tid >> 1;
        int nh = (tid & 1) * 8;
        float Acn[8], h[8];
        #pragma unroll
        for (int n = 0; n < 8; ++n) {
            Acn[n] = -__expf(A_log[c * N_STATE + nh + n]);
            h[n]   = 0.f;
        }
        float Dc = Dvec[c];
        for (int l = 0; l < L_DIM; ++l) {
            float dl = delta_s[l * C_DIM + c];
            float ul = u_s[l * U_STR + c];
            float db = dl * ul;
            float acc = 0.f;
            #pragma unroll
            for (int n = 0; n < 8; ++n) {
                float Bn = proj_s[l * P_STR + RT + nh + n];            // broadcast read
                float Cn = proj_s[l * P_STR + RT + N_STATE + nh + n];  // broadcast read
                h[n] = __expf(dl * Acn[n]) * h[n] + db * Bn;
                acc += h[n] * Cn;
            }
            acc += __shfl_xor(acc, 1, 32);           // pair-reduce the two state halves
            if ((tid & 1) == 0)
                u_s[l * U_STR + c] = acc + ul * Dc;  // y overwrites consumed u slot
        }
    }
    __syncthreads();

    // ---- Phase 5: writeback (h-branch overwrites, v-branch accumulates) ----
    for (int idx = tid; idx < L_DIM * C_DIM; idx += 256) {
        int c = idx >> 7, l = idx & 127;
        float v = u_s[l * U_STR + c];
        int o = base + c * HW + l * l_stride;
        if (accumulate) out[o] += v; else out[o] = v;
    }
}

extern "C" void kernel_launch(void* const* d_in, const int* in_sizes, int n_in,
                              void* d_out, int out_size, void* d_ws, size_t ws_size,
                              hipStream_t stream) {
    const float* x     = (const float*)d_in[0];
    const float* A_log = (const float*)d_in[1];
    const float* Dvec  = (const float*)d_in[2];
    const float* xpw   = (const float*)d_in[3];
    const float* dtw   = (const float*)d_in[4];
    const float* dtb   = (const float*)d_in[5];
    float* out = (float*)d_out;

    const int B = in_sizes[0] / (C_DIM * HW);   // 4
    dim3 grid(B * 128), block(256);

    // Horizontal scan: sequences (b,h), l = w (contiguous). Overwrites out.
    mamba_seq_kernel<<<grid, block, 0, stream>>>(x, A_log, Dvec, xpw, dtw, dtb, out,
                                                 /*inner_mul=*/128, /*l_stride=*/1,
                                                 /*accumulate=*/0);
    // Vertical scan: sequences (b,w), l = h (stride W, L2-resident). Accumulates.
    mamba_seq_kernel<<<grid, block, 0, stream>>>(x, A_log, Dvec, xpw, dtw, dtb, out,
                                                 /*inner_mul=*/1, /*l_stride=*/128,
                                                 /*accumulate=*/1);
}